// Embedding_29523605193133
// MI455X (gfx1250) — compile-verified
//
#include <hip/hip_runtime.h>

typedef __attribute__((ext_vector_type(2))) float v2f;
typedef __attribute__((ext_vector_type(4))) float f4;
typedef __attribute__((ext_vector_type(8))) float v8f;

#define EMBED_DIM 768
#define SEQ_LEN   2048
#define N_ROWS    (32 * 2048)
#define LN_EPS    1e-5f
// ln(10000) / 384  (since 2*i/768 = i/384)
#define PE_LN_COEF 0.02398526138535465f

__global__ __launch_bounds__(256) void embed_pe_ln_kernel(
    const int*   __restrict__ x,
    const float* __restrict__ table,
    const float* __restrict__ gamma,
    const float* __restrict__ beta,
    float*       __restrict__ out)
{
    const int lane        = threadIdx.x & 31;
    const int wave        = threadIdx.x >> 5;
    const int wave_global = blockIdx.x * 8 + wave;
    const int nwaves      = gridDim.x * 8;

    // gamma/beta are tiny (3 KB each) and shared by every row this wave touches:
    // load once per wave, keep in VGPRs.
    f4 g[6], bta[6];
#pragma unroll
    for (int k = 0; k < 6; ++k) {
        const int off = lane * 4 + k * 128;
        g[k]   = *(const f4*)(gamma + off);
        bta[k] = *(const f4*)(beta  + off);
    }

    for (int r = wave_global; r < N_ROWS; r += nwaves) {
        const int idx = x[r];                         // wave-uniform -> scalar load
        const float* __restrict__ src = table + (size_t)idx * EMBED_DIM;

        // Prefetch next row's table line through L2 (global_prefetch_b8).
        // 32 lanes x 96 B spacing covers the full 3 KB row's cachelines.
        if (r + nwaves < N_ROWS) {
            const int idx_next = x[r + nwaves];
            __builtin_prefetch(table + (size_t)idx_next * EMBED_DIM + lane * 24, 0, 3);
        }

        // Coalesced gather of one 768-f32 row: 6 x global_load_b128 per lane.
        f4 h[6];
#pragma unroll
        for (int k = 0; k < 6; ++k)
            h[k] = *(const f4*)(src + lane * 4 + k * 128);

        // Add positional encoding, accumulate per-lane sum / sum-of-squares.
        const float pf = (float)(r & (SEQ_LEN - 1));
        float sum = 0.0f, sq = 0.0f;
#pragma unroll
        for (int k = 0; k < 6; ++k) {
            const int ibase = lane * 4 + k * 128;
#pragma unroll
            for (int j = 0; j < 4; ++j) {
                const float ang = pf * __expf(-(float)(ibase + j) * PE_LN_COEF);
                const float pe  = (j & 1) ? __cosf(ang) : __sinf(ang);
                const float v   = h[k][j] + pe;
                h[k][j] = v;
                sum += v;
                sq  += v * v;
            }
        }

        // ---- Cross-lane LayerNorm reduction via V_WMMA_F32_16X16X4_F32 ----
        // A layout (f32 16x4): lane l, VGPR0 holds A[l%16][2*(l/16)]; VGPR1 zeroed.
        // With B = all-ones, D[m][n] = A[m][0] + A[m][2] = sum_m + sum_{m+16}.
        // Each lane then adds its 8 D-VGPRs (rows m=lane-group) and one
        // shfl_xor(16) completes the 32-lane reduction on every lane.
        v2f a_sum; a_sum.x = sum; a_sum.y = 0.0f;
        v2f a_sq;  a_sq.x  = sq;  a_sq.y  = 0.0f;
        v2f ones;  ones.x  = 1.0f; ones.y = 1.0f;
        v8f cz = {};
        v8f dsum = __builtin_amdgcn_wmma_f32_16x16x4_f32(
            false, a_sum, false, ones, (short)0, cz, false, false);
        v8f dsq  = __builtin_amdgcn_wmma_f32_16x16x4_f32(
            false, a_sq,  false, ones, (short)0, cz, false, false);
        float ts = 0.0f, tq = 0.0f;
#pragma unroll
        for (int e = 0; e < 8; ++e) { ts += dsum[e]; tq += dsq[e]; }
        ts += __shfl_xor(ts, 16, 32);
        tq += __shfl_xor(tq, 16, 32);

        const float mu   = ts * (1.0f / (float)EMBED_DIM);
        const float var  = tq * (1.0f / (float)EMBED_DIM) - mu * mu;
        const float rstd = rsqrtf(var + LN_EPS);

        // Normalize + affine, stream out with non-temporal hint so the 201 MB
        // output does not evict the L2-resident 154 MB embedding table.
        float* __restrict__ dst = out + (size_t)r * EMBED_DIM;
#pragma unroll
        for (int k = 0; k < 6; ++k) {
            f4 o;
#pragma unroll
            for (int j = 0; j < 4; ++j)
                o[j] = (h[k][j] - mu) * rstd * g[k][j] + bta[k][j];
            __builtin_nontemporal_store(o, (f4*)(dst + lane * 4 + k * 128));
        }
    }
}

extern "C" void kernel_launch(void* const* d_in, const int* in_sizes, int n_in,
                              void* d_out, int out_size, void* d_ws, size_t ws_size,
                              hipStream_t stream) {
    (void)in_sizes; (void)n_in; (void)out_size; (void)d_ws; (void)ws_size;
    const int*   x     = (const int*)d_in[0];
    const float* table = (const float*)d_in[1];
    const float* gamma = (const float*)d_in[2];
    const float* beta  = (const float*)d_in[3];
    float*       out   = (float*)d_out;

    // 65536 rows, 8 waves/block, 4 rows/wave -> 2048 blocks, exact coverage
    // (EXEC stays all-1s, as WMMA requires).
    dim3 grid(2048), block(256);
    embed_pe_ln_kernel<<<grid, block, 0, stream>>>(x, table, gamma, beta, out);
}